// PDE_N9_52106543235553
// MI455X (gfx1250) — compile-verified
//
#include <hip/hip_runtime.h>

typedef __attribute__((ext_vector_type(4))) int   v4i;
typedef __attribute__((ext_vector_type(4))) float v4f;

// ---------------------------------------------------------------------------
// Phase 2 hot path (4 edges/lane, NO tail handling -> branch-free body):
//   msg[dst[e]] += w[e] * g[src[e]]
//
// nVec = nE/4 vector-groups. Edge streams are read exactly once (192 MB) ->
// non-temporal loads keep L2 reserved for the hot 2 MB g[] (gather) and
// msg[] (atomic scatter) arrays. Atomic adds are relaxed/agent so the
// backend emits native no-return global_atomic_add_f32 (STOREcnt only).
// ---------------------------------------------------------------------------
__global__ __launch_bounds__(256) void pde_edge_scatter_vec(
    const v4i* __restrict__ src4,
    const v4i* __restrict__ dst4,
    const v4f* __restrict__ w4,
    const float* __restrict__ g,
    float*       __restrict__ msg,
    int nVec)
{
    int t = blockIdx.x * blockDim.x + threadIdx.x;
    if (t < nVec) {
        // 16B coalesced non-temporal loads: one wave pulls 512B per stream.
        v4i s  = __builtin_nontemporal_load(src4 + t);
        v4i d  = __builtin_nontemporal_load(dst4 + t);
        v4f ww = __builtin_nontemporal_load(w4   + t);

        // Gathers from g[] stay regular-temporal: g is 2 MB, L2-resident.
        float m0 = ww.x * g[s.x];
        float m1 = ww.y * g[s.y];
        float m2 = ww.z * g[s.z];
        float m3 = ww.w * g[s.w];

        __hip_atomic_fetch_add(&msg[d.x], m0, __ATOMIC_RELAXED, __HIP_MEMORY_SCOPE_AGENT);
        __hip_atomic_fetch_add(&msg[d.y], m1, __ATOMIC_RELAXED, __HIP_MEMORY_SCOPE_AGENT);
        __hip_atomic_fetch_add(&msg[d.z], m2, __ATOMIC_RELAXED, __HIP_MEMORY_SCOPE_AGENT);
        __hip_atomic_fetch_add(&msg[d.w], m3, __ATOMIC_RELAXED, __HIP_MEMORY_SCOPE_AGENT);
    }
}

// ---------------------------------------------------------------------------
// Phase 2 tail (one edge per lane). Launched only when nE % 4 != 0 (never
// for the reference shapes: 16M % 4 == 0). Deterministic: the launch
// decision depends only on nE.
// ---------------------------------------------------------------------------
__global__ __launch_bounds__(64) void pde_edge_scatter_tail(
    const int*   __restrict__ src,
    const int*   __restrict__ dst,
    const float* __restrict__ w,
    const float* __restrict__ g,
    float*       __restrict__ msg,
    int start, int nE)
{
    int e = start + (int)(blockIdx.x * blockDim.x + threadIdx.x);
    if (e < nE) {
        float me = w[e] * g[src[e]];
        __hip_atomic_fetch_add(&msg[dst[e]], me,
                               __ATOMIC_RELAXED, __HIP_MEMORY_SCOPE_AGENT);
    }
}

// ---------------------------------------------------------------------------
// Phase 1 (per node): g[i] = relu(voltage[i]) * type_params[neuron_type[i]]
//                     msg[i] = 0   (re-zero scratch every call: deterministic)
// ---------------------------------------------------------------------------
__global__ __launch_bounds__(256) void pde_node_pre(
    const float* __restrict__ voltage,
    const int*   __restrict__ ntype,
    const float* __restrict__ type_params,
    float* __restrict__ g,
    float* __restrict__ msg,
    int n)
{
    int i = blockIdx.x * blockDim.x + threadIdx.x;
    if (i < n) {
        float v = voltage[i];
        float r = v > 0.0f ? v : 0.0f;
        g[i]   = r * type_params[ntype[i]];
        msg[i] = 0.0f;
    }
}

// ---------------------------------------------------------------------------
// Phase 3 (per node): dv[i] = (-v + msg + stimulus + V_rest) / tau
// ---------------------------------------------------------------------------
__global__ __launch_bounds__(256) void pde_node_post(
    const float* __restrict__ voltage,
    const float* __restrict__ stimulus,
    const float* __restrict__ vrest,
    const float* __restrict__ tau,
    const float* __restrict__ msg,
    float*       __restrict__ out,
    int n)
{
    int i = blockIdx.x * blockDim.x + threadIdx.x;
    if (i < n) {
        out[i] = (-voltage[i] + msg[i] + stimulus[i] + vrest[i]) / tau[i];
    }
}

// ---------------------------------------------------------------------------
// Launch. Inputs (setup_inputs order):
//   0 voltage f32[N], 1 stimulus f32[N], 2 neuron_type i32[N],
//   3 edge_index i32[2*E] (row0=src, row1=dst), 4 w f32[E],
//   5 V_i_rest f32[N], 6 tau_i f32[N], 7 type_params f32[64]
// Output: dv f32[N].  Scratch: g f32[N] | msg f32[N]  (4 MB of d_ws).
// ---------------------------------------------------------------------------
extern "C" void kernel_launch(void* const* d_in, const int* in_sizes, int n_in,
                              void* d_out, int out_size, void* d_ws, size_t ws_size,
                              hipStream_t stream)
{
    const float* voltage  = (const float*)d_in[0];
    const float* stimulus = (const float*)d_in[1];
    const int*   ntype    = (const int*)  d_in[2];
    const int*   eidx     = (const int*)  d_in[3];
    const float* w        = (const float*)d_in[4];
    const float* vrest    = (const float*)d_in[5];
    const float* tau      = (const float*)d_in[6];
    const float* tparams  = (const float*)d_in[7];

    const int nN = in_sizes[0];
    const int nE = in_sizes[4];

    const int* src = eidx;        // edge_index[0, :]
    const int* dst = eidx + nE;   // edge_index[1, :]

    float* g   = (float*)d_ws;    // nN floats
    float* msg = g + nN;          // nN floats

    float* out = (float*)d_out;

    const int BLK = 256;

    // Phase 1: per-node precompute + msg zeroing
    int nodeBlocks = (nN + BLK - 1) / BLK;
    pde_node_pre<<<nodeBlocks, BLK, 0, stream>>>(voltage, ntype, tparams, g, msg, nN);

    // Phase 2: edge scatter, 4 edges per lane, branch-free hot kernel
    int nVec = nE / 4;
    if (nVec > 0) {
        int edgeBlocks = (nVec + BLK - 1) / BLK;
        pde_edge_scatter_vec<<<edgeBlocks, BLK, 0, stream>>>(
            (const v4i*)src, (const v4i*)dst, (const v4f*)w, g, msg, nVec);
    }
    int tailStart = nVec * 4;
    if (tailStart < nE) {  // only when nE % 4 != 0 (never for reference shapes)
        pde_edge_scatter_tail<<<1, 64, 0, stream>>>(src, dst, w, g, msg, tailStart, nE);
    }

    // Phase 3: per-node finalize
    pde_node_post<<<nodeBlocks, BLK, 0, stream>>>(voltage, stimulus, vrest, tau, msg, out, nN);
}